// EncoderProcessorDecoderGNN_78958678770242
// MI455X (gfx1250) — compile-verified
//
#include <hip/hip_runtime.h>

typedef float v2f __attribute__((ext_vector_type(2)));
typedef float v8f __attribute__((ext_vector_type(8)));

#define HIDC 128

// ---------------------------------------------------------------- utility ops
__global__ __launch_bounds__(256) void zero_f32_k(float* __restrict__ p, int n) {
    int i = blockIdx.x * 256 + threadIdx.x;
    if (i < n) p[i] = 0.0f;
}

__global__ __launch_bounds__(256) void deg_count_k(const int* __restrict__ dst,
                                                   float* __restrict__ deg, int E) {
    int e = blockIdx.x * 256 + threadIdx.x;
    if (e < E)
        __hip_atomic_fetch_add(&deg[dst[e]], 1.0f,
                               __ATOMIC_RELAXED, __HIP_MEMORY_SCOPE_AGENT);
}

// deg -> dinv in place. Self-loop contributes +1, so deg+1 > 0 always.
__global__ __launch_bounds__(256) void deg_to_dinv_k(float* __restrict__ deg, int n) {
    int i = blockIdx.x * 256 + threadIdx.x;
    if (i < n) deg[i] = rsqrtf(deg[i] + 1.0f);
}

__global__ __launch_bounds__(256) void norm_k(const int* __restrict__ src,
                                              const int* __restrict__ dst,
                                              const float* __restrict__ dinv,
                                              float* __restrict__ nrm, int E) {
    int e = blockIdx.x * 256 + threadIdx.x;
    if (e < E) nrm[e] = dinv[src[e]] * dinv[dst[e]];
}

// ------------------------------------------------------- fp32 WMMA GEMM tile
// OUT[N x 128] = (A_RELU ? relu(A) : A)[N x K] * W[K x 128]  (+ bias if ADD_BIAS)
// If HAS_ACC, also write ACC[row][col] = bias[col] + out * dinv[row]^2
// (fused scatter-accumulator init incl. self-loop term; ACC may alias A — each
//  block writes only the 16 rows it alone reads, after all its A reads).
// grid.x = N/16 row tiles; block = 64 threads = 2 waves; wave w covers cols [64w, 64w+64)
template <int A_RELU, int ADD_BIAS, bool HAS_ACC>
__global__ __launch_bounds__(64) void gemm_wmma_k(const float* A, int lda,
                                                  const float* __restrict__ W,
                                                  const float* __restrict__ bias,
                                                  float* __restrict__ OUT,
                                                  float* ACC,
                                                  const float* __restrict__ dinv,
                                                  int K) {
    const int lane    = threadIdx.x & 31;
    const int wave    = threadIdx.x >> 5;        // 0..1
    const int row0    = blockIdx.x * 16;
    const int colbase = wave * 64;
    const int m       = lane & 15;               // A row within tile / B-D col within tile
    const int kg      = (lane >> 4) << 1;        // K-pair select: 0 (lanes 0-15) or 2 (16-31)

    v8f acc[4];
    #pragma unroll
    for (int t = 0; t < 4; ++t) acc[t] = (v8f){};

    const float* arow = A + (size_t)(row0 + m) * lda + kg;

    for (int k = 0; k < K; k += 4) {
        // A frag: 16x4 fp32 -> 2 VGPRs/lane: VGPR0=K(kg), VGPR1=K(kg+1)
        v2f a;
        float a0 = arow[k + 0];
        float a1 = arow[k + 1];
        a.x = A_RELU ? fmaxf(a0, 0.0f) : a0;
        a.y = A_RELU ? fmaxf(a1, 0.0f) : a1;
        // B frags: 4x16 fp32 per N-tile; lane holds col m, rows kg / kg+1
        const float* wk = W + (size_t)(k + kg) * HIDC + m;
        #pragma unroll
        for (int t = 0; t < 4; ++t) {
            v2f b;
            b.x = wk[colbase + t * 16];
            b.y = wk[HIDC + colbase + t * 16];
            acc[t] = __builtin_amdgcn_wmma_f32_16x16x4_f32(
                false, a, false, b, (short)0, acc[t], false, false);
        }
    }

    const int rowoff = (lane >> 4) << 3;         // 0 or 8
    float d2[8];
    if (HAS_ACC) {
        #pragma unroll
        for (int r = 0; r < 8; ++r) {
            float di = dinv[row0 + rowoff + r];
            d2[r] = di * di;
        }
    }
    #pragma unroll
    for (int t = 0; t < 4; ++t) {
        int col = colbase + t * 16 + m;
        float bv = bias[col];
        #pragma unroll
        for (int r = 0; r < 8; ++r) {
            size_t off = (size_t)(row0 + rowoff + r) * HIDC + col;
            float v = acc[t][r];
            if (ADD_BIAS) v += bv;
            OUT[off] = v;
            if (HAS_ACC) ACC[off] = bv + v * d2[r];   // bias + self-loop contribution
        }
    }
}

// ---------------------------------------------------- edge scatter: 1 wave / edge
// acc[dst] += hw[src] * norm[e]  (128 ch: 4 per lane, float4 gather from L2-resident hw)
__global__ __launch_bounds__(256) void scatter_k(const int* __restrict__ esrc,
                                                 const int* __restrict__ edst,
                                                 const float* __restrict__ nrm,
                                                 const float* __restrict__ hw,
                                                 float* __restrict__ acc, int E) {
    int e    = (int)((blockIdx.x * 256 + threadIdx.x) >> 5);
    int lane = threadIdx.x & 31;
    if (e >= E) return;
    int   s = esrc[e];
    int   d = edst[e];
    float w = nrm[e];
    float4 v = ((const float4*)(hw + (size_t)s * HIDC))[lane];
    float* dp = acc + (size_t)d * HIDC + lane * 4;
    __hip_atomic_fetch_add(dp + 0, v.x * w, __ATOMIC_RELAXED, __HIP_MEMORY_SCOPE_AGENT);
    __hip_atomic_fetch_add(dp + 1, v.y * w, __ATOMIC_RELAXED, __HIP_MEMORY_SCOPE_AGENT);
    __hip_atomic_fetch_add(dp + 2, v.z * w, __ATOMIC_RELAXED, __HIP_MEMORY_SCOPE_AGENT);
    __hip_atomic_fetch_add(dp + 3, v.w * w, __ATOMIC_RELAXED, __HIP_MEMORY_SCOPE_AGENT);
}

// ------------------------------------------------- decoder (relu fused on load)
__global__ __launch_bounds__(256) void decoder_k(const float* __restrict__ h,
                                                 const float* __restrict__ W,
                                                 const float* __restrict__ b,
                                                 float* __restrict__ out, int N) {
    int i = blockIdx.x * 256 + threadIdx.x;
    if (i >= N) return;
    const float* hr = h + (size_t)i * HIDC;
    float a0 = b[0], a1 = b[1], a2 = b[2];
    #pragma unroll 4
    for (int k = 0; k < HIDC; ++k) {
        float hv = fmaxf(hr[k], 0.0f);   // relu deferred from last conv layer
        a0 += hv * W[k * 3 + 0];
        a1 += hv * W[k * 3 + 1];
        a2 += hv * W[k * 3 + 2];
    }
    out[(size_t)i * 3 + 0] = a0;
    out[(size_t)i * 3 + 1] = a1;
    out[(size_t)i * 3 + 2] = a2;
}

extern "C" void kernel_launch(void* const* d_in, const int* in_sizes, int n_in,
                              void* d_out, int out_size, void* d_ws, size_t ws_size,
                              hipStream_t stream) {
    const float* x      = (const float*)d_in[0];
    const int*   ei     = (const int*)d_in[1];
    const float* enc_W  = (const float*)d_in[2];
    const float* enc_b  = (const float*)d_in[3];
    const float* conv_W = (const float*)d_in[4];
    const float* conv_b = (const float*)d_in[5];
    const float* dec_W  = (const float*)d_in[6];
    const float* dec_b  = (const float*)d_in[7];

    const int N = in_sizes[0] / 16;   // IN_CH = 16
    const int E = in_sizes[1] / 2;    // edge_index is [2, E]
    const int* esrc = ei;
    const int* edst = ei + E;

    // workspace carve (floats): h | hw | dinv | norm  (~110 MB total)
    float* h    = (float*)d_ws;
    float* hw   = h    + (size_t)N * HIDC;
    float* dinv = hw   + (size_t)N * HIDC;
    float* nrm  = dinv + N;

    const int gN    = (N + 255) / 256;
    const int gE    = (E + 255) / 256;
    const int tiles = (N + 15) / 16;          // N = 100000 -> 6250, exact
    const int gS    = (E + 7) / 8;            // 8 waves per 256-thread block

    // degree -> dinv -> edge norms (recomputed each call; ws is not re-zeroed)
    zero_f32_k   <<<gN, 256, 0, stream>>>(dinv, N);
    deg_count_k  <<<gE, 256, 0, stream>>>(edst, dinv, E);
    deg_to_dinv_k<<<gN, 256, 0, stream>>>(dinv, N);
    norm_k       <<<gE, 256, 0, stream>>>(esrc, edst, dinv, nrm, E);

    // encoder: h = x @ enc_W + enc_b   (relu deferred to first conv A-load)
    gemm_wmma_k<0, 1, false><<<tiles, 64, 0, stream>>>(x, 16, enc_W, enc_b, h,
                                                       nullptr, nullptr, 16);

    for (int l = 0; l < 4; ++l) {
        // hw = relu(h) @ conv_W[l]; simultaneously h = conv_b[l] + hw*dinv^2
        gemm_wmma_k<1, 0, true><<<tiles, 64, 0, stream>>>(
            h, HIDC,
            conv_W + (size_t)l * HIDC * HIDC,
            conv_b + (size_t)l * HIDC,
            hw, h, dinv, HIDC);
        // h[dst] += hw[src] * norm  (relu deferred to next consumer)
        scatter_k<<<gS, 256, 0, stream>>>(esrc, edst, nrm, hw, h, E);
    }

    decoder_k<<<gN, 256, 0, stream>>>(h, dec_W, dec_b, (float*)d_out, N);
}